// SyllableProcessor_2731599200860
// MI455X (gfx1250) — compile-verified
//
#include <hip/hip_runtime.h>
#include <math.h>

typedef __attribute__((ext_vector_type(2))) float v2f;
typedef __attribute__((ext_vector_type(4))) float v4f;
typedef __attribute__((ext_vector_type(8))) float v8f;

#define HD   256           // hidden size H
#define K2   512           // 2*H (GEMM K)
#define MT   64            // rows (characters) per block
#define MS   4             // M-subtiles of 16 rows (register blocking)
#define LN_EPS 1e-5f

__global__ __launch_bounds__(256) void syllable_fused_kernel(
    const float* __restrict__ char_emb,     // [B, Lc, H]
    const long long* __restrict__ align_,   // [B, Lc] int64
    const float* __restrict__ dec,          // [B, S, H]
    const float* __restrict__ W1,           // [2H, H] row-major (K,N)
    const float* __restrict__ b1,           // [H]
    const float* __restrict__ W2,           // [2H, H]
    const float* __restrict__ b2,           // [H]
    const float* __restrict__ gamma,        // [H]
    const float* __restrict__ beta,         // [H]
    float* __restrict__ out,                // [B, Lc, H]
    int Lc, int S)
{
    // 128 KB combined tile: [64][512] = [char | tok] -> [char | aligned].
    // After GEMM2 the first 64 KB is reused for feats [64][256].
    __shared__ float lds_comb[MT * K2];
    __shared__ float red_s[MT][4];
    __shared__ float red_q[MT][4];
    __shared__ float mu_s[MT];
    __shared__ float rs_s[MT];
    float* lds_feat = lds_comb;             // alias (combined is dead after GEMM2)

    const int tid  = threadIdx.x;
    const int lane = tid & 31;
    const int wave = tid >> 5;

    const int tiles_per_b = Lc / MT;
    const int bidx = blockIdx.x / tiles_per_b;
    const int mt   = blockIdx.x % tiles_per_b;
    const long long row0 = (long long)bidx * Lc + (long long)mt * MT;

    // ---------------- stage char tile + gathered token reprs into LDS ----------------
    // 4 threads per row, 64 consecutive floats each.
    {
        const int r  = tid >> 2;            // 0..63
        const int c0 = (tid & 3) << 6;      // 0,64,128,192
        const float* src = char_emb + (row0 + r) * HD + c0;
        float* dstc = &lds_comb[r * K2 + c0];
        #pragma unroll
        for (int i = 0; i < 64; i += 4)
            *(v4f*)(dstc + i) = *(const v4f*)(src + i);

        long long ta = align_[row0 + r];
        float* dstt = &lds_comb[r * K2 + HD + c0];
        if (ta < 0) {
            #pragma unroll
            for (int i = 0; i < 64; i += 4)
                *(v4f*)(dstt + i) = (v4f){0.f, 0.f, 0.f, 0.f};
        } else {
            long long si = ta > (long long)(S - 1) ? (long long)(S - 1) : ta;
            const float* ts = dec + ((long long)bidx * S + si) * HD + c0;
            #pragma unroll
            for (int i = 0; i < 64; i += 4)
                *(v4f*)(dstt + i) = *(const v4f*)(ts + i);
        }
    }
    __syncthreads();

    // WMMA fragment coordinates (16x16x4 f32 layout):
    //  A (16x4): lanes 0-15 hold K=0,1 ; lanes 16-31 hold K=2,3
    //  B (4x16): lanes = col N,  VGPR0/1 = K pair per half-wave
    //  C (16x16): VGPR j, lanes 0-15 -> M=j, lanes 16-31 -> M=j+8; N = lane%16
    const int mrow = lane & 15;
    const int koff = (lane >> 4) << 1;      // 0 or 2
    const int ncol = lane & 15;
    const int mhi  = (lane >> 4) << 3;      // +0 or +8 row offset in C layout
    const int nb0  = wave * 32;             // this wave's 32-column slab

    // ---------------- GEMM1: [char|tok] x W1 + b1 ----------------
    {
        v8f acc[MS][2];
        #pragma unroll
        for (int t = 0; t < 2; ++t) {
            const float bv = b1[nb0 + t * 16 + ncol];
            #pragma unroll
            for (int s = 0; s < MS; ++s)
                #pragma unroll
                for (int j = 0; j < 8; ++j) acc[s][t][j] = bv;
        }

        for (int k = 0; k < K2; k += 4) {
            v2f bf[2];
            #pragma unroll
            for (int t = 0; t < 2; ++t) {
                const int n = nb0 + t * 16 + ncol;
                bf[t].x = W1[(k + koff    ) * HD + n];
                bf[t].y = W1[(k + koff + 1) * HD + n];
            }
            #pragma unroll
            for (int s = 0; s < MS; ++s) {
                v2f a = *(const v2f*)&lds_comb[(s * 16 + mrow) * K2 + k + koff];
                #pragma unroll
                for (int t = 0; t < 2; ++t)
                    acc[s][t] = __builtin_amdgcn_wmma_f32_16x16x4_f32(
                        false, a, false, bf[t], (short)0, acc[s][t], false, false);
            }
        }
        __syncthreads();   // all reads of the tok half are done

        // write 'aligned' into the second half -> lds_comb becomes [char | aligned]
        #pragma unroll
        for (int s = 0; s < MS; ++s)
            #pragma unroll
            for (int t = 0; t < 2; ++t)
                #pragma unroll
                for (int j = 0; j < 8; ++j)
                    lds_comb[(s * 16 + j + mhi) * K2 + HD + nb0 + t * 16 + ncol] = acc[s][t][j];
        __syncthreads();
    }

    // ---------------- GEMM2: [char|aligned] x W2 + b2 ----------------
    {
        v8f acc[MS][2];
        #pragma unroll
        for (int t = 0; t < 2; ++t) {
            const float bv = b2[nb0 + t * 16 + ncol];
            #pragma unroll
            for (int s = 0; s < MS; ++s)
                #pragma unroll
                for (int j = 0; j < 8; ++j) acc[s][t][j] = bv;
        }

        for (int k = 0; k < K2; k += 4) {
            v2f bf[2];
            #pragma unroll
            for (int t = 0; t < 2; ++t) {
                const int n = nb0 + t * 16 + ncol;
                bf[t].x = W2[(k + koff    ) * HD + n];
                bf[t].y = W2[(k + koff + 1) * HD + n];
            }
            #pragma unroll
            for (int s = 0; s < MS; ++s) {
                v2f a = *(const v2f*)&lds_comb[(s * 16 + mrow) * K2 + k + koff];
                #pragma unroll
                for (int t = 0; t < 2; ++t)
                    acc[s][t] = __builtin_amdgcn_wmma_f32_16x16x4_f32(
                        false, a, false, bf[t], (short)0, acc[s][t], false, false);
            }
        }
        __syncthreads();   // combined tile is dead; its storage becomes feats

        // park feats in (aliased) LDS for row-wise LayerNorm
        #pragma unroll
        for (int s = 0; s < MS; ++s)
            #pragma unroll
            for (int t = 0; t < 2; ++t)
                #pragma unroll
                for (int j = 0; j < 8; ++j)
                    lds_feat[(s * 16 + j + mhi) * HD + nb0 + t * 16 + ncol] = acc[s][t][j];
        __syncthreads();
    }

    // ---------------- LayerNorm stats (two-stage: 4 partials per row) ----------------
    {
        const int r = tid >> 2, q = tid & 3;
        float s = 0.f, qq = 0.f;
        #pragma unroll
        for (int i = 0; i < 64; ++i) {
            float x = lds_feat[r * HD + q * 64 + i];
            s += x; qq += x * x;
        }
        red_s[r][q] = s;
        red_q[r][q] = qq;
    }
    __syncthreads();
    if (tid < MT) {
        float s = 0.f, q = 0.f;
        #pragma unroll
        for (int c = 0; c < 4; ++c) { s += red_s[tid][c]; q += red_q[tid][c]; }
        const float mu  = s * (1.0f / HD);
        const float var = q * (1.0f / HD) - mu * mu;
        mu_s[tid] = mu;
        rs_s[tid] = rsqrtf(var + LN_EPS);
    }
    __syncthreads();

    // ---------------- normalize + exact GELU + residual + store ----------------
    {
        const int r  = tid >> 2;
        const int c0 = (tid & 3) << 6;
        const float mu = mu_s[r];
        const float rs = rs_s[r];
        const float* ce = char_emb + (row0 + r) * HD + c0;
        float* op = out + (row0 + r) * HD + c0;
        #pragma unroll
        for (int i = 0; i < 64; ++i) {
            float x = (lds_feat[r * HD + c0 + i] - mu) * rs * gamma[c0 + i] + beta[c0 + i];
            float g = 0.5f * x * (1.0f + erff(x * 0.70710678118654752f)); // exact GELU
            op[i] = ce[i] + g;
        }
    }
}

extern "C" void kernel_launch(void* const* d_in, const int* in_sizes, int n_in,
                              void* d_out, int out_size, void* d_ws, size_t ws_size,
                              hipStream_t stream) {
    const float*     char_emb = (const float*)d_in[0];
    const long long* align_   = (const long long*)d_in[1];   // int64 in reference
    const float*     dec      = (const float*)d_in[2];
    const float*     W1       = (const float*)d_in[3];
    const float*     b1       = (const float*)d_in[4];
    const float*     W2       = (const float*)d_in[5];
    const float*     b2       = (const float*)d_in[6];
    const float*     gamma    = (const float*)d_in[7];
    const float*     beta     = (const float*)d_in[8];
    float* out = (float*)d_out;

    const int B  = 8;                               // per reference setup
    const int total_rows = in_sizes[1];             // B * Lc
    const int Lc = total_rows / B;
    const int S  = in_sizes[2] / (B * HD);

    const int nblocks = total_rows / MT;            // 2048
    syllable_fused_kernel<<<dim3(nblocks), dim3(256), 0, stream>>>(
        char_emb, align_, dec, W1, b1, W2, b2, gamma, beta, out, Lc, S);
}